// LRD2_61057255080031
// MI455X (gfx1250) — compile-verified
//
#include <hip/hip_runtime.h>

typedef __attribute__((ext_vector_type(2))) float v2f;
typedef __attribute__((ext_vector_type(8))) float v8f;

constexpr int N    = 25;            // support per episode
constexpr int NWAY = 5;
constexpr int QN   = 75;            // queries per episode
constexpr int D    = 512;
constexpr int KC   = 64;            // K-chunk staged in LDS
constexpr int NCH  = D / KC;        // 8 chunks
constexpr int SROWS = 32;           // padded support rows
constexpr int QROWS = 80;           // padded query rows
constexpr int ROWS  = SROWS + QROWS;        // 112 staged rows
constexpr int RSTR  = KC + 4;               // 68 floats: bank = 4*row + col -> conflict-free
constexpr int BUFSZ = ROWS * RSTR;          // 7616 floats per stage buffer
constexpr int THREADS = 224;                // 7 waves
constexpr int LD_PER_T = ROWS * (KC / 4) / THREADS;  // 1792/224 = 8 async b128 loads/thread/chunk

__global__ __launch_bounds__(THREADS)
void lrd2_dual_irls_kernel(const float* __restrict__ zs,
                           const float* __restrict__ zq,
                           const float* __restrict__ lamp,
                           float* __restrict__ out)
{
    __shared__ float smem[2 * BUFSZ];   // 60928 B: double-buffered stage; buf0 reused for results

    const int tid  = threadIdx.x;
    const int wv   = tid >> 5;
    const int lane = tid & 31;
    const int b    = blockIdx.x;
    const float lam = lamp[0];

    const size_t zsb = (size_t)b * N  * D;
    const size_t zqb = (size_t)b * QN * D;

    // ---- Per-thread async-load slots: 8 x b128 per chunk, uniform across all threads
    // so per-wave ASYNCcnt increments by exactly 8 per chunk.
    // Buffer row l: 0..31 = support (pad rows alias row 0), 32..111 = query (pad alias row 0).
    const float* gsrc[LD_PER_T];
    unsigned     ldst[LD_PER_T];
    #pragma unroll
    for (int i = 0; i < LD_PER_T; ++i) {
        int t = tid + i * THREADS;
        int l = t >> 4;              // staged row (16 float4 per row)
        int c = (t & 15) << 2;       // float column within chunk
        const float* base;
        if (l < SROWS) { int g = (l < N)  ? l  : 0; base = zs + zsb + (size_t)g * D; }
        else { int lq = l - SROWS; int g = (lq < QN) ? lq : 0; base = zq + zqb + (size_t)g * D; }
        gsrc[i] = base + c;
        ldst[i] = (unsigned)((l * RSTR + c) * 4);   // byte offset within a stage buffer
    }
    const unsigned lds0 = (unsigned)(uintptr_t)smem;

    auto issue = [&](int bufsel, int kb) {
        #pragma unroll
        for (int i = 0; i < LD_PER_T; ++i) {
            unsigned     d = lds0 + (unsigned)bufsel * (BUFSZ * 4u) + ldst[i];
            const float* s = gsrc[i] + kb;
            // CDNA5 async global->LDS DMA, tracked by ASYNCcnt (GV addressing mode)
            asm volatile("global_load_async_to_lds_b128 %0, %1, off"
                         :: "v"(d), "v"(s) : "memory");
        }
    };

    v8f acc0 = {};   // this wave's 16x16 tile vs support cols 0..15
    v8f acc1 = {};   // vs support cols 16..31
    const int lr = lane & 15;
    const int kh = (lane >> 4) << 1;   // K sub-offset per ISA A/B fragment layout

    issue(0, 0);                        // prefetch chunk 0
    for (int ch = 0; ch < NCH; ++ch) {
        if (ch + 1 < NCH) {
            issue((ch + 1) & 1, (ch + 1) * KC);           // prefetch next chunk (8 more)
            asm volatile("s_wait_asynccnt 8" ::: "memory"); // in-order: prev batch done
        } else {
            asm volatile("s_wait_asynccnt 0" ::: "memory");
        }
        __syncthreads();   // all waves' chunk-ch data visible in LDS

        const float* buf = smem + (ch & 1) * BUFSZ;
        // Wave w: M-tiles 0,1 = support rows (G = zs zs^T), tiles 2..6 = query rows (Kqs)
        const float* Ab = (wv < 2) ? buf + (wv * 16 + lr) * RSTR
                                   : buf + (SROWS + (wv - 2) * 16 + lr) * RSTR;
        const float* B0 = buf + lr * RSTR;          // support rows 0..15 (B = Xs^T)
        const float* B1 = buf + (16 + lr) * RSTR;   // support rows 16..31
        #pragma unroll 4
        for (int kk = 0; kk < KC; kk += 4) {
            v2f a  = *(const v2f*)(Ab + kk + kh);
            v2f b0 = *(const v2f*)(B0 + kk + kh);
            v2f b1 = *(const v2f*)(B1 + kk + kh);
            acc0 = __builtin_amdgcn_wmma_f32_16x16x4_f32(false, a, false, b0, (short)0, acc0, false, false);
            acc1 = __builtin_amdgcn_wmma_f32_16x16x4_f32(false, a, false, b1, (short)0, acc1, false, false);
        }
        __syncthreads();   // everyone done reading buf before it is refilled (ch+2)
    }

    // ---- Result regions overlaid on stage buffer 0 (7231 floats <= 7616)
    float* XXt  = smem;                            // [32][33]  X X^T (incl. +1 bias)
    float* Kqs  = smem + 32 * 33;                  // [80][33]  Xq Xs^T (incl. +1 bias)
    float* Asol = smem + 32 * 33 + 80 * 33;        // [5][25][27] augmented solve matrices
    float* dual = smem + 32 * 33 + 80 * 33 + 5 * 25 * 27;  // [5][32]

    {   // C/D layout: VGPR i -> row i (lanes 0-15) / row 8+i (lanes 16-31), col = lane&15
        const int rt = (lane >> 4) * 8;
        const int cn = lane & 15;
        #pragma unroll
        for (int i = 0; i < 8; ++i) {
            float e0 = acc0[i] + 1.0f;   // + ones(N,N) from the bias column
            float e1 = acc1[i] + 1.0f;
            if (wv < 2) {
                int row = wv * 16 + rt + i;
                XXt[row * 33 + cn]      = e0;
                XXt[row * 33 + 16 + cn] = e1;
            } else {
                int row = (wv - 2) * 16 + rt + i;
                Kqs[row * 33 + cn]      = e0;
                Kqs[row * 33 + 16 + cn] = e1;
            }
        }
    }
    __syncthreads();

    // ---- Phase 2: dual-space IRLS, wave c = class c (waves 5,6 idle but hit barriers)
    const bool act = (wv < NWAY);
    const int  cls = act ? wv : 0;
    float* Ac = Asol + cls * (25 * 27);
    float* dc = dual + cls * 32;
    const int r = lane;
    const float ylab = (r >= 5 * cls && r < 5 * cls + 5) ? 1.0f : 0.0f;
    float eta = 0.0f;

    for (int it = 0; it < 3; ++it) {
        if (act && r < N) {  // lane r builds augmented row r
            float mu = 1.0f / (1.0f + __expf(-eta));
            float s  = mu * (1.0f - mu);
            float zr = eta + (ylab - mu) / s;
            for (int j = 0; j < N; ++j)
                Ac[r * 27 + j] = XXt[r * 33 + j] + ((j == r) ? lam / s : 0.0f);
            Ac[r * 27 + 25] = zr;
        }
        __syncthreads();
        // Forward elimination (SPD -> no pivoting); lane j owns column j (col 25 = rhs)
        for (int k = 0; k < N; ++k) {
            if (act) {
                float inv = 1.0f / Ac[k * 27 + k];
                float akj = (lane < 26) ? Ac[k * 27 + lane] : 0.0f;
                for (int r2 = k + 1; r2 < N; ++r2) {
                    float f = Ac[r2 * 27 + k] * inv;
                    if (lane < 26 && lane > k) Ac[r2 * 27 + lane] -= f * akj;
                }
            }
            __syncthreads();
        }
        // Back substitution; lane r owns rhs row r
        for (int k = N - 1; k >= 0; --k) {
            if (act) {
                float xk = Ac[k * 27 + 25] / Ac[k * 27 + k];
                if (lane == 0) dc[k] = xk;
                if (lane < k) Ac[lane * 27 + 25] -= Ac[lane * 27 + k] * xk;
            }
            __syncthreads();
        }
        if (act && it < 2) {  // eta = X w = (X X^T) dual
            float e = 0.0f;
            for (int j = 0; j < N; ++j) e += XXt[r * 33 + j] * dc[j];
            eta = e;
        }
        __syncthreads();
    }

    // scores[:, cls] = Kqs @ dual  (Kqs already includes bias +1)
    if (act) {
        for (int q = lane; q < QN; q += 32) {
            float sc = 0.0f;
            for (int j = 0; j < N; ++j) sc += Kqs[q * 33 + j] * dc[j];
            out[((size_t)b * QN + q) * NWAY + cls] = sc;
        }
    }
}

extern "C" void kernel_launch(void* const* d_in, const int* in_sizes, int n_in,
                              void* d_out, int out_size, void* d_ws, size_t ws_size,
                              hipStream_t stream) {
    const float* zs  = (const float*)d_in[0];
    const float* zq  = (const float*)d_in[1];
    const float* lam = (const float*)d_in[2];
    float* out = (float*)d_out;
    const int B = in_sizes[0] / (N * D);   // 2048
    lrd2_dual_irls_kernel<<<B, THREADS, 0, stream>>>(zs, zq, lam, out);
}